// FastfoodTransform_52295521796385
// MI455X (gfx1250) — compile-verified
//
#include <hip/hip_runtime.h>
#include <math.h>
#include <stdint.h>

typedef __attribute__((ext_vector_type(2))) float v2f;
typedef __attribute__((ext_vector_type(8))) float v8f;
typedef __attribute__((ext_vector_type(4))) int   v4i;

#define LLBITS 23
#define LLSZ   (1 << LLBITS)   // 8388608
#define DOUT   8000000
#define N1     4096            // contiguous chunk (16^3), done LAST
#define N2     2048            // column length (16^2 * 8), done FIRST (stride N1)
#define PAC    4               // columns per pass-A block
#define PAPITCH 2052           // padded LDS row pitch (2052*4 bytes, 16B aligned)

#define AS1 __attribute__((address_space(1)))
#define AS3 __attribute__((address_space(3)))

// H16[m][k] = (-1)^popc(m & k)
__device__ __forceinline__ float hsign(int m, int k) {
  return (__popc(m & k) & 1) ? -1.0f : 1.0f;
}

// One 16x16x16 Hadamard tile: 16 independent 16-point transforms in LDS.
// Transform n touches buf[base0 + n*nstride + k*kstride], k=0..15.
// V_WMMA_F32_16X16X4_F32 chained 4x (K=16). Exact f32 math (A is +-1).
__device__ __forceinline__ void h16_tile(float* buf, int base0, int nstride,
                                         int kstride) {
  const int lane = threadIdx.x & 31;
  const int n    = lane & 15;
  const int hi   = lane >> 4;
  const int cb   = base0 + n * nstride;
  const int k0   = 2 * hi;

  v2f b0, b1, b2, b3, a0, a1, a2, a3;
  b0.x = buf[cb + (k0 + 0)  * kstride];  b0.y = buf[cb + (k0 + 1)  * kstride];
  b1.x = buf[cb + (k0 + 4)  * kstride];  b1.y = buf[cb + (k0 + 5)  * kstride];
  b2.x = buf[cb + (k0 + 8)  * kstride];  b2.y = buf[cb + (k0 + 9)  * kstride];
  b3.x = buf[cb + (k0 + 12) * kstride];  b3.y = buf[cb + (k0 + 13) * kstride];

  a0.x = hsign(n, k0 + 0);   a0.y = hsign(n, k0 + 1);
  a1.x = hsign(n, k0 + 4);   a1.y = hsign(n, k0 + 5);
  a2.x = hsign(n, k0 + 8);   a2.y = hsign(n, k0 + 9);
  a3.x = hsign(n, k0 + 12);  a3.y = hsign(n, k0 + 13);

  v8f c = {};
  c = __builtin_amdgcn_wmma_f32_16x16x4_f32(false, a0, false, b0, (short)0, c, false, false);
  c = __builtin_amdgcn_wmma_f32_16x16x4_f32(false, a1, false, b1, (short)0, c, false, false);
  c = __builtin_amdgcn_wmma_f32_16x16x4_f32(false, a2, false, b2, (short)0, c, false, false);
  c = __builtin_amdgcn_wmma_f32_16x16x4_f32(false, a3, false, b3, (short)0, c, false, false);

#pragma unroll
  for (int r = 0; r < 8; ++r)
    buf[cb + (r + 8 * hi) * kstride] = c[r];
}

// One radix-16 FWHT stage (stride s; s==1 or multiple of 16) over a
// contiguous LDS region of 'len' floats. Tiles are element-disjoint, striped
// over waves (scalar wave index -> uniform branching around WMMA).
__device__ __forceinline__ void fwht16_stage(float* region, int len, int s) {
  const int wave   = __builtin_amdgcn_readfirstlane(threadIdx.x) >> 5;
  const int nwaves = blockDim.x >> 5;
  const int ntiles = len >> 8;
  for (int g = wave; g < ntiles; g += nwaves) {
    const int t0 = g << 4;
    if (s == 1) {
      h16_tile(region, t0 << 4, 16, 1);
    } else {
      const int base0 = (t0 / s) * (s << 4) + (t0 % s);
      h16_tile(region, base0, 1, s);
    }
  }
}

__device__ __forceinline__ void radix2_stage(float* region, int len, int s) {
  const int pairs = len >> 1;
  for (int p = threadIdx.x; p < pairs; p += blockDim.x) {
    const int i = ((p & ~(s - 1)) << 1) | (p & (s - 1));
    const float a = region[i], b = region[i + s];
    region[i]     = a + b;
    region[i + s] = a - b;
  }
}

// ---- kernel 1: w = FWHT_1024(BB[:1024] * x) --------------------------------
__global__ __launch_bounds__(256) void k_small_fwht(const float* __restrict__ x,
                                                    const float* __restrict__ BB,
                                                    float* __restrict__ w) {
  __shared__ __align__(16) float buf[1024];
  for (int i = threadIdx.x; i < 1024; i += blockDim.x) buf[i] = x[i] * BB[i];
  __syncthreads();
  fwht16_stage(buf, 1024, 1);   __syncthreads();
  fwht16_stage(buf, 1024, 16);  __syncthreads();
  radix2_stage(buf, 1024, 256); __syncthreads();
  radix2_stage(buf, 1024, 512); __syncthreads();
  for (int i = threadIdx.x; i < 1024; i += blockDim.x) w[i] = buf[i];
}

// ---- kernel 2a/2b: deterministic reduction of sum(GG^2) --------------------
__global__ __launch_bounds__(256) void k_gg2(const float* __restrict__ GG,
                                             float* __restrict__ partial) {
  __shared__ float red[256];
  const float4* g4 = (const float4*)(GG + blockIdx.x * 8192);
  float s = 0.0f;
  for (int j = threadIdx.x; j < 2048; j += 256) {
    const float4 v = g4[j];
    s += v.x * v.x + v.y * v.y + v.z * v.z + v.w * v.w;
  }
  red[threadIdx.x] = s;
  __syncthreads();
  for (int off = 128; off > 0; off >>= 1) {
    if (threadIdx.x < off) red[threadIdx.x] += red[threadIdx.x + off];
    __syncthreads();
  }
  if (threadIdx.x == 0) partial[blockIdx.x] = red[0];
}

__global__ __launch_bounds__(256) void k_scale(const float* __restrict__ partial,
                                               float* __restrict__ scalep) {
  __shared__ float red[256];
  float s = 0.0f;
  for (int j = threadIdx.x; j < 1024; j += 256) s += partial[j];
  red[threadIdx.x] = s;
  __syncthreads();
  for (int off = 128; off > 0; off >>= 1) {
    if (threadIdx.x < off) red[threadIdx.x] += red[threadIdx.x + off];
    __syncthreads();
  }
  // out = mul5[:D] / (sqrt(LL*S)*sqrt(D/LL)) = mul5[:D] / sqrt(S*D)
  if (threadIdx.x == 0) scalep[0] = 1.0f / sqrtf(red[0] * (float)DOUT);
}

// ---- kernel 3 (pass A): gather + FWHT over HIGH 11 bits (stride 4096) ------
__global__ __launch_bounds__(256) void k_passA(const float* __restrict__ GG,
                                               const int* __restrict__ Pi,
                                               const float* __restrict__ w,
                                               float* __restrict__ tmp) {
  __shared__ __align__(16) float buf[PAC * PAPITCH];
  const int c0 = blockIdx.x * PAC;

  // gather 4 columns x 2048 rows; b128 loads; transpose into padded LDS rows
  for (int i2 = threadIdx.x; i2 < N2; i2 += 256) {
    const int g = i2 * N1 + c0;
    __builtin_prefetch(Pi + g + 256 * N1, 0, 1);
    __builtin_prefetch(GG + g + 256 * N1, 0, 1);
    const int4   p4 = *(const int4*)(Pi + g);
    const float4 v4 = *(const float4*)(GG + g);
    // mul_2 is 1024-periodic: mul_2[Pi] = w[Pi & 1023]
    buf[0 * PAPITCH + i2] = w[p4.x & 1023] * v4.x;
    buf[1 * PAPITCH + i2] = w[p4.y & 1023] * v4.y;
    buf[2 * PAPITCH + i2] = w[p4.z & 1023] * v4.z;
    buf[3 * PAPITCH + i2] = w[p4.w & 1023] * v4.w;
  }
  __syncthreads();

  for (int cc = 0; cc < PAC; ++cc) fwht16_stage(buf + cc * PAPITCH, N2, 1);
  __syncthreads();
  for (int cc = 0; cc < PAC; ++cc) fwht16_stage(buf + cc * PAPITCH, N2, 16);
  __syncthreads();
  for (int cc = 0; cc < PAC; ++cc) radix2_stage(buf + cc * PAPITCH, N2, 256);
  __syncthreads();
  for (int cc = 0; cc < PAC; ++cc) radix2_stage(buf + cc * PAPITCH, N2, 512);
  __syncthreads();
  for (int cc = 0; cc < PAC; ++cc) radix2_stage(buf + cc * PAPITCH, N2, 1024);
  __syncthreads();

  for (int i2 = threadIdx.x; i2 < N2; i2 += 256) {
    float4 v;
    v.x = buf[0 * PAPITCH + i2];
    v.y = buf[1 * PAPITCH + i2];
    v.z = buf[2 * PAPITCH + i2];
    v.w = buf[3 * PAPITCH + i2];
    *(float4*)(tmp + i2 * N1 + c0) = v;   // 16B stores, L2-resident buffer
  }
}

// ---- kernel 4 (pass B): FWHT over LOW 12 bits (contiguous) + scaled store --
__global__ __launch_bounds__(256) void k_passB(const float* __restrict__ tmp,
                                               const float* __restrict__ scalep,
                                               float* __restrict__ out) {
  __shared__ __align__(16) float buf[N1];
  const int base = blockIdx.x * N1;

#if defined(__has_builtin) && __has_builtin(__builtin_amdgcn_global_load_async_to_lds_b128)
  // L2 -> LDS without a VGPR round-trip (ASYNCcnt path).
  // Builtin expects typed v4i pointers: global src (AS1), LDS dst (AS3).
  for (int i = threadIdx.x * 4; i < N1; i += 256 * 4) {
    __builtin_amdgcn_global_load_async_to_lds_b128(
        (AS1 v4i*)(uintptr_t)(tmp + base + i),   // flat==global address bits
        (AS3 v4i*)(buf + i),                     // generic->LDS addrspacecast
        0, 0);
  }
  asm volatile("s_wait_asynccnt 0" ::: "memory");
#else
  for (int i = threadIdx.x * 4; i < N1; i += 256 * 4)
    *(float4*)(buf + i) = *(const float4*)(tmp + base + i);
#endif
  __syncthreads();

  fwht16_stage(buf, N1, 1);    __syncthreads();
  fwht16_stage(buf, N1, 16);   __syncthreads();
  fwht16_stage(buf, N1, 256);  __syncthreads();

  const float scale = scalep[0];
  const int   g40   = base >> 2;
  for (int i4 = threadIdx.x; i4 < (N1 >> 2); i4 += 256) {
    if (g40 + i4 < (DOUT >> 2)) {           // D % 4 == 0
      float4 v = *(const float4*)(buf + 4 * i4);
      v.x *= scale; v.y *= scale; v.z *= scale; v.w *= scale;
      *(float4*)(out + base + 4 * i4) = v;  // fully coalesced HBM store
    }
  }
}

extern "C" void kernel_launch(void* const* d_in, const int* in_sizes, int n_in,
                              void* d_out, int out_size, void* d_ws, size_t ws_size,
                              hipStream_t stream) {
  (void)in_sizes; (void)n_in; (void)out_size; (void)ws_size;
  const float* x  = (const float*)d_in[0];
  const float* BB = (const float*)d_in[1];
  const float* GG = (const float*)d_in[2];
  const int*   Pi = (const int*)d_in[3];
  float* out = (float*)d_out;

  char*  ws      = (char*)d_ws;
  float* w       = (float*)(ws);            // 1024 f32
  float* partial = (float*)(ws + 4096);     // 1024 f32
  float* scalep  = (float*)(ws + 8192);     // 1 f32
  float* tmp     = (float*)(ws + 16384);    // LLSZ f32 (32 MiB, L2-resident)

  k_small_fwht<<<1, 256, 0, stream>>>(x, BB, w);
  k_gg2<<<LLSZ / 8192, 256, 0, stream>>>(GG, partial);
  k_scale<<<1, 256, 0, stream>>>(partial, scalep);
  k_passA<<<N1 / PAC, 256, 0, stream>>>(GG, Pi, w, tmp);
  // only chunks that intersect [0, D) are needed
  k_passB<<<(DOUT + N1 - 1) / N1, 256, 0, stream>>>(tmp, scalep, out);
}